// Sampler_15169824489889
// MI455X (gfx1250) — compile-verified
//
#include <hip/hip_runtime.h>
#include <hip/hip_bf16.h>
#include <stdint.h>

#define TPB 512
#define TILE_ELEMS 2048           // 512 threads * 4 floats = 8KB tile
#define NBINS 2048
#define CAND_CAP 512
#define PREFIX 128
#define SAMP_EPS 1e-9f
#define F32_LOWEST -3.4028234663852886e+38f

// ---------- CDNA5 async global->LDS copy (ASYNCcnt path) --------------------
#if __has_builtin(__builtin_amdgcn_global_load_async_to_lds_b128)
#define USE_ASYNC_LDS 1
#else
#define USE_ASYNC_LDS 0
#endif

typedef int v4i_t __attribute__((ext_vector_type(4)));

__device__ __forceinline__ void async_copy_b128(const float* g, float* l) {
#if USE_ASYNC_LDS
  v4i_t* gv = (v4i_t*)const_cast<float*>(g);   // generic, non-const
  __builtin_amdgcn_global_load_async_to_lds_b128(
      (__attribute__((address_space(1))) v4i_t*)gv,
      (__attribute__((address_space(3))) v4i_t*)l, 0, 0);
#else
  *(float4*)l = *(const float4*)g;
#endif
}

#if USE_ASYNC_LDS
# if __has_builtin(__builtin_amdgcn_s_wait_asynccnt)
#  define WAIT_ASYNC(n) __builtin_amdgcn_s_wait_asynccnt(n)
# else
#  define WAIT_ASYNC(n) asm volatile("s_wait_asynccnt " #n ::: "memory")
# endif
#else
# define WAIT_ASYNC(n) ((void)0)
#endif

// monotone float -> uint32 key (ascending), probs are always >= 0
__device__ __forceinline__ uint32_t ord_f32(float f) {
  uint32_t u = __float_as_uint(f);
  return (u & 0x80000000u) ? ~u : (u | 0x80000000u);
}

// ===========================================================================
// K1: per-row max, sum(exp), and candidate threshold (LDS histogram).
// Streams the row twice through async-LDS double buffering (2nd pass from L2).
// ===========================================================================
__global__ __launch_bounds__(TPB) void sampler_k1_stats(
    const float* __restrict__ logits, const float* __restrict__ temps,
    float* __restrict__ mOut, float* __restrict__ sOut,
    float* __restrict__ thrOut, int* __restrict__ candCount, int V) {
  __shared__ __align__(16) float tile[2][TILE_ELEMS];
  __shared__ uint32_t hist[NBINS];
  __shared__ float red[TPB];
  __shared__ float shm;

  const int row = blockIdx.x;
  const int tid = threadIdx.x;
  const float t = temps[row];
  const float* rowp = logits + (size_t)row * (size_t)V;
  const int nT = (V + TILE_ELEMS - 1) / TILE_ELEMS;

  for (int b = tid; b < NBINS; b += TPB) hist[b] = 0u;
  __syncthreads();

  // ---- phase 1: max + histogram (only x >= 0 tail is histogrammed) ----
  float lmax = __uint_as_float(0xff800000u);  // -inf
  {
    int b0 = tid * 4;
    if (b0 + 4 <= V) async_copy_b128(rowp + b0, &tile[0][tid * 4]);
  }
  for (int tt = 0; tt < nT; ++tt) {
    const int cur = tt & 1, nxt = cur ^ 1;
    const bool hasNext = (tt + 1 < nT);
    const int nb = (tt + 1) * TILE_ELEMS + tid * 4;
    if (hasNext && nb + 4 <= V) async_copy_b128(rowp + nb, &tile[nxt][tid * 4]);
    if (hasNext) { WAIT_ASYNC(1); } else { WAIT_ASYNC(0); }
    const int base = tt * TILE_ELEMS + tid * 4;
    if (base + 4 <= V) {
      float4 v = *(const float4*)&tile[cur][tid * 4];
      float xs[4] = {v.x / t, v.y / t, v.z / t, v.w / t};
#pragma unroll
      for (int k = 0; k < 4; ++k) {
        float x = xs[k];
        lmax = fmaxf(lmax, x);
        int bin = (int)((x + 16.0f) * 64.0f);
        if (bin > NBINS - 1) bin = NBINS - 1;
        if (bin >= NBINS / 2) atomicAdd(&hist[bin], 1u);
      }
    }
  }
  __syncthreads();
  red[tid] = lmax;
  __syncthreads();
  for (int s2 = TPB / 2; s2 > 0; s2 >>= 1) {
    if (tid < s2) red[tid] = fmaxf(red[tid], red[tid + s2]);
    __syncthreads();
  }
  if (tid == 0) shm = red[0];
  __syncthreads();
  const float m = shm;
  __syncthreads();

  // ---- phase 2: sum(exp(x - m)) (row re-read hits L2) ----
  float lsum = 0.0f;
  {
    int b0 = tid * 4;
    if (b0 + 4 <= V) async_copy_b128(rowp + b0, &tile[0][tid * 4]);
  }
  for (int tt = 0; tt < nT; ++tt) {
    const int cur = tt & 1, nxt = cur ^ 1;
    const bool hasNext = (tt + 1 < nT);
    const int nb = (tt + 1) * TILE_ELEMS + tid * 4;
    if (hasNext && nb + 4 <= V) async_copy_b128(rowp + nb, &tile[nxt][tid * 4]);
    if (hasNext) { WAIT_ASYNC(1); } else { WAIT_ASYNC(0); }
    const int base = tt * TILE_ELEMS + tid * 4;
    if (base + 4 <= V) {
      float4 v = *(const float4*)&tile[cur][tid * 4];
      lsum += expf(v.x / t - m) + expf(v.y / t - m) +
              expf(v.z / t - m) + expf(v.w / t - m);
    }
  }
  __syncthreads();
  red[tid] = lsum;
  __syncthreads();
  for (int s2 = TPB / 2; s2 > 0; s2 >>= 1) {
    if (tid < s2) red[tid] += red[tid + s2];
    __syncthreads();
  }
  if (tid == 0) {
    // threshold: highest bin edge whose suffix count >= 144 (slack for
    // float-binning edges); take one extra bin of safety.
    uint32_t acc = 0;
    float thr = 0.0f;
    for (int b = NBINS - 1; b >= NBINS / 2; --b) {
      acc += hist[b];
      if (acc >= 144u) { thr = (float)(b - 1) * (1.0f / 64.0f) - 16.0f; break; }
    }
    mOut[row] = m;
    sOut[row] = red[0];
    thrOut[row] = thr;
    candCount[row] = 0;
  }
}

// ===========================================================================
// K2: write log_probs (clamped), compact top candidates (prob-key packed).
// ===========================================================================
__global__ __launch_bounds__(TPB) void sampler_k2_lp(
    const float* __restrict__ logits, const float* __restrict__ temps,
    const float* __restrict__ mArr, const float* __restrict__ sArr,
    const float* __restrict__ thrArr, int* __restrict__ candCount,
    unsigned long long* __restrict__ cand, float* __restrict__ lp, int V) {
  __shared__ __align__(16) float tile[2][TILE_ELEMS];
  const int row = blockIdx.x;
  const int tid = threadIdx.x;
  const float t = temps[row], m = mArr[row], s = sArr[row], thr = thrArr[row];
  const float* rowp = logits + (size_t)row * (size_t)V;
  float* lprow = lp ? lp + (size_t)row * (size_t)V : nullptr;
  const int nT = (V + TILE_ELEMS - 1) / TILE_ELEMS;

  {
    int b0 = tid * 4;
    if (b0 + 4 <= V) async_copy_b128(rowp + b0, &tile[0][tid * 4]);
  }
  for (int tt = 0; tt < nT; ++tt) {
    const int cur = tt & 1, nxt = cur ^ 1;
    const bool hasNext = (tt + 1 < nT);
    const int nb = (tt + 1) * TILE_ELEMS + tid * 4;
    if (hasNext && nb + 4 <= V) async_copy_b128(rowp + nb, &tile[nxt][tid * 4]);
    if (hasNext) { WAIT_ASYNC(1); } else { WAIT_ASYNC(0); }
    const int base = tt * TILE_ELEMS + tid * 4;
    if (base + 4 <= V) {
      float4 v = *(const float4*)&tile[cur][tid * 4];
      float xs[4] = {v.x / t, v.y / t, v.z / t, v.w / t};
      float lpv[4];
#pragma unroll
      for (int k = 0; k < 4; ++k) {
        float x = xs[k];
        float p = expf(x - m) / s;                 // matches reference rounding path
        lpv[k] = fmaxf(logf(p), F32_LOWEST);       // clip(min=finfo.min)
        if (x >= thr) {
          int slot = atomicAdd(&candCount[row], 1);
          if (slot < CAND_CAP)
            cand[(size_t)row * CAND_CAP + slot] =
                ((unsigned long long)ord_f32(p) << 32) | (uint32_t)(base + k);
        }
      }
      if (lprow) {
        float4 o; o.x = lpv[0]; o.y = lpv[1]; o.z = lpv[2]; o.w = lpv[3];
        *(float4*)&lprow[base] = o;
      }
    }
  }
}

// ===========================================================================
// K3: bitonic-sort candidates (desc by prob, desc by index on ties == reversed
// stable argsort), apply top-k/top-p/min-p, gumbel-max over all V ranks.
// ===========================================================================
__global__ __launch_bounds__(TPB) void sampler_k3_sample(
    const unsigned long long* __restrict__ cand, const int* __restrict__ candCount,
    const int* __restrict__ top_ks, const float* __restrict__ top_ps,
    const float* __restrict__ min_ps, const int* __restrict__ positions,
    const int* __restrict__ seeds, int* __restrict__ selRank,
    float* __restrict__ outTok, int V) {
  __shared__ unsigned long long skey[CAND_CAP];
  __shared__ float pv[PREFIX], qv[PREFIX], ov[PREFIX];
  __shared__ float rv[TPB];
  __shared__ int rj[TPB];
  __shared__ int sh_L;
  __shared__ float sh_q0, sh_Z, sh_base;

  const int row = blockIdx.x;
  const int tid = threadIdx.x;
  int cnt = candCount[row];
  if (cnt > CAND_CAP) cnt = CAND_CAP;

  skey[tid] = (tid < cnt) ? cand[(size_t)row * CAND_CAP + tid] : 0ull;
  __syncthreads();

  // bitonic sort, descending
  for (int k = 2; k <= CAND_CAP; k <<= 1) {
    for (int j = k >> 1; j > 0; j >>= 1) {
      int ixj = tid ^ j;
      if (ixj > tid) {
        unsigned long long a = skey[tid], b = skey[ixj];
        bool descBlk = ((tid & k) == 0);
        if (descBlk ? (a < b) : (a > b)) { skey[tid] = b; skey[ixj] = a; }
      }
      __syncthreads();
    }
  }

  const int L = (cnt < PREFIX) ? cnt : PREFIX;
  if (tid < PREFIX) {
    float p = 0.0f;
    if (tid < L) p = __uint_as_float((uint32_t)(skey[tid] >> 32) ^ 0x80000000u);
    pv[tid] = p;
  }
  __syncthreads();

  if (tid == 0) {
    const int topk = top_ks[row];
    const float topp = top_ps[row], minp = min_ps[row];
    float cum = 0.0f;
    for (int j = 0; j < L; ++j) {                 // top-k then top-p (pre-mask cumsum)
      float pj = pv[j];
      cum += pj;
      float q = (j < topk) ? pj : 0.0f;
      if ((cum - q) > topp) q = 0.0f;
      qv[j] = q;
    }
    float q0 = (L > 0) ? qv[0] : 0.0f;            // row max of filtered array
    float mthr = q0 * minp;
    int nk = 0;
    float sk = 0.0f;
    for (int j = 0; j < L; ++j) {                 // min-p, kept stats
      float q = qv[j];
      if (q < mthr) q = 0.0f;
      qv[j] = q;
      if (q != 0.0f) { nk++; sk += expf(q - q0); }
    }
    const float e0 = expf(-q0);
    const float Z = sk + (float)(V - nk) * e0;    // softmax denom over all V ranks
    sh_q0 = q0; sh_Z = Z; sh_base = logf(e0 / Z + SAMP_EPS); sh_L = L;
  }
  __syncthreads();
  if (tid < PREFIX) {
    float o = sh_base;
    if (tid < sh_L && qv[tid] != 0.0f)
      o = logf(expf(qv[tid] - sh_q0) / sh_Z + SAMP_EPS);
    ov[tid] = o;
  }
  __syncthreads();

  // seeded gumbel-max over ranks 0..V-1 (pure compute, no memory)
  const uint32_t ss =
      ((uint32_t)seeds[row] * 19349663u) ^ ((uint32_t)positions[row] * 73856093u);
  const float base_o = sh_base;
  float bestV = __uint_as_float(0xff800000u);
  int bestJ = 0;
  for (int j = tid; j < V; j += TPB) {
    uint32_t h = ss * 805306457u ^ ((uint32_t)j * 479001599u);
    float u = (float)(h & 0x00FFFFFFu) * (1.0f / 16777216.0f);
    float g = -logf(-logf(u + SAMP_EPS) + SAMP_EPS);
    float val = ((j < PREFIX) ? ov[j] : base_o) + g;
    if (val > bestV) { bestV = val; bestJ = j; }  // first-tie wins within thread
  }
  rv[tid] = bestV;
  rj[tid] = bestJ;
  __syncthreads();
  for (int s2 = TPB / 2; s2 > 0; s2 >>= 1) {
    if (tid < s2) {
      float v2 = rv[tid + s2];
      int j2 = rj[tid + s2];
      if (v2 > rv[tid] || (v2 == rv[tid] && j2 < rj[tid])) { rv[tid] = v2; rj[tid] = j2; }
    }
    __syncthreads();
  }
  if (tid == 0) {
    int r = rj[0];
    if (r < sh_L) {                               // rank inside known prefix
      outTok[row] = (float)(int)(skey[r] & 0xFFFFFFFFull);
      selRank[row] = -1;
    } else {
      selRank[row] = r;                           // needs order-statistic select
    }
  }
}

// ===========================================================================
// K4: radix-select the vocab id at sorted rank r (desc by prob bits, ties by
// descending index). 4x8-bit passes + tie resolve, all data L2-resident.
// ===========================================================================
__global__ __launch_bounds__(TPB) void sampler_k4_select(
    const float* __restrict__ logits, const float* __restrict__ temps,
    const float* __restrict__ mArr, const float* __restrict__ sArr,
    const int* __restrict__ selRank, float* __restrict__ outTok, int V) {
  __shared__ uint32_t hist[256];
  __shared__ int sh_byte, sh_rr;
  __shared__ uint32_t matches[256];
  __shared__ int mcount;

  const int row = blockIdx.x;
  const int tid = threadIdx.x;
  const int r = selRank[row];
  if (r < 0) return;  // uniform per block

  const float t = temps[row], m = mArr[row], s = sArr[row];
  const float* rowp = logits + (size_t)row * (size_t)V;
  uint32_t prefix = 0, mask = 0;
  int rr = r;

  for (int pass = 0; pass < 4; ++pass) {
    const int shift = 24 - 8 * pass;
    if (tid < 256) hist[tid] = 0u;
    __syncthreads();
    for (int i = tid; i < V; i += TPB) {
      float x = rowp[i] / t;
      uint32_t key = ord_f32(expf(x - m) / s);
      if ((key & mask) == prefix) atomicAdd(&hist[(key >> shift) & 0xFFu], 1u);
    }
    __syncthreads();
    if (tid == 0) {
      uint32_t acc = 0;
      int bsel = 0, nrr = rr;
      for (int b = 255; b >= 0; --b) {
        uint32_t c = hist[b];
        if (acc + c > (uint32_t)rr) { bsel = b; nrr = rr - (int)acc; break; }
        acc += c;
      }
      sh_byte = bsel;
      sh_rr = nrr;
    }
    __syncthreads();
    prefix |= ((uint32_t)sh_byte) << shift;
    mask |= 0xFFu << shift;
    rr = sh_rr;
    __syncthreads();
  }

  if (tid == 0) mcount = 0;
  __syncthreads();
  for (int i = tid; i < V; i += TPB) {
    float x = rowp[i] / t;
    uint32_t key = ord_f32(expf(x - m) / s);
    if (key == prefix) {
      int slot = atomicAdd(&mcount, 1);
      if (slot < 256) matches[slot] = (uint32_t)i;
    }
  }
  __syncthreads();
  if (tid == 0) {
    int n = (mcount < 256) ? mcount : 256;
    int tok = 0;
    for (int a = 0; a < n; ++a) {   // rank within tie group: descending index
      int c = 0;
      for (int b = 0; b < n; ++b)
        if (matches[b] > matches[a]) c++;
      if (c == rr) { tok = (int)matches[a]; break; }
    }
    outTok[row] = (float)tok;
  }
}

// ===========================================================================
extern "C" void kernel_launch(void* const* d_in, const int* in_sizes, int n_in,
                              void* d_out, int out_size, void* d_ws, size_t ws_size,
                              hipStream_t stream) {
  const float* logits = (const float*)d_in[0];
  const float* temps = (const float*)d_in[1];
  const int* top_ks = (const int*)d_in[2];
  const float* top_ps = (const float*)d_in[3];
  const float* min_ps = (const float*)d_in[4];
  const int* positions = (const int*)d_in[5];
  const int* seeds = (const int*)d_in[6];

  const int B = in_sizes[2];
  const int V = in_sizes[0] / B;

  float* out_tok = (float*)d_out;
  float* out_lp =
      ((long long)out_size >= (long long)B + (long long)B * (long long)V)
          ? out_tok + B : nullptr;

  char* ws = (char*)d_ws;
  float* mArr = (float*)ws;    ws += (size_t)B * sizeof(float);
  float* sArr = (float*)ws;    ws += (size_t)B * sizeof(float);
  float* thrArr = (float*)ws;  ws += (size_t)B * sizeof(float);
  int* candCount = (int*)ws;   ws += (size_t)B * sizeof(int);
  int* selRank = (int*)ws;     ws += (size_t)B * sizeof(int);
  ws = (char*)(((uintptr_t)ws + 15u) & ~(uintptr_t)15u);
  unsigned long long* cand = (unsigned long long*)ws;
  (void)ws_size; (void)n_in;

  dim3 grid(B), blk(TPB);
  sampler_k1_stats<<<grid, blk, 0, stream>>>(logits, temps, mArr, sArr, thrArr,
                                             candCount, V);
  sampler_k2_lp<<<grid, blk, 0, stream>>>(logits, temps, mArr, sArr, thrArr,
                                          candCount, cand, out_lp, V);
  sampler_k3_sample<<<grid, blk, 0, stream>>>(cand, candCount, top_ks, top_ps,
                                              min_ps, positions, seeds, selRank,
                                              out_tok, V);
  sampler_k4_select<<<grid, blk, 0, stream>>>(logits, temps, mArr, sArr, selRank,
                                              out_tok, V);
}